// VectorQuantizer_23167053594886
// MI455X (gfx1250) — compile-verified
//
#include <hip/hip_runtime.h>

typedef __attribute__((ext_vector_type(2))) float v2f;
typedef __attribute__((ext_vector_type(8))) float v8f;
typedef __attribute__((ext_vector_type(4))) int   v4i;

typedef __attribute__((address_space(1))) v4i* gv4i_p;   // global int4*
typedef __attribute__((address_space(3))) v4i* lv4i_p;   // LDS int4*

#define K_CODES 1024
#define DIMS 32
#define N_ROWS 131072
#define QN (N_ROWS * DIMS)      // 4194304 floats of q output
#define LOSS_OFF QN
#define IDX_OFF (QN + 1)
#define ROWS_PER_WAVE 16
#define WAVES_PER_BLOCK 8
#define ROWS_PER_BLOCK (ROWS_PER_WAVE * WAVES_PER_BLOCK)

// Zero the single loss accumulator before the main kernel (harness does not
// re-clear d_out between graph replays).
__global__ void vq_zero_loss(float* __restrict__ out) {
    out[LOSS_OFF] = 0.0f;
}

__global__ __launch_bounds__(256) void vq_argmin_wmma(const float* __restrict__ x,
                                                      const float* __restrict__ W,
                                                      float* __restrict__ out) {
    // Stage the whole codebook (128 KB) + half-norms (4 KB) in LDS.
    __shared__ float Wl[K_CODES * DIMS];
    __shared__ float Nl[K_CODES];

    const int t = threadIdx.x;

#if __has_builtin(__builtin_amdgcn_global_load_async_to_lds_b128)
    // CDNA5 async copy: memory -> LDS directly, no VGPR round trip, ASYNCcnt-tracked.
    {
#pragma unroll
        for (int i = 0; i < (K_CODES * DIMS * 4 / 16) / 256; ++i) {
            const long off = (long)(t + i * 256) * 16;  // byte offset, 16B granules
            gv4i_p gsrc = (gv4i_p)(uintptr_t)((const char*)W + off);
            lv4i_p ldst = (lv4i_p)(unsigned)(uintptr_t)((char*)Wl + off);
            __builtin_amdgcn_global_load_async_to_lds_b128(gsrc, ldst, 0, 0);
        }
    }
#define VQ_ASYNC_STAGE 1
#else
    {
        const float4* __restrict__ src = (const float4*)W;
        float4* __restrict__ dst = (float4*)Wl;
#pragma unroll
        for (int i = 0; i < (K_CODES * DIMS / 4) / 256; ++i)
            dst[t + i * 256] = src[t + i * 256];
    }
#define VQ_ASYNC_STAGE 0
#endif

    // Half-norms 0.5*||w_k||^2 computed from global W (L2-hot), overlapping the
    // async LDS copy above. 4 codes per thread.
#pragma unroll
    for (int cc = 0; cc < 4; ++cc) {
        const int code = t * 4 + cc;
        const float4* __restrict__ wr = (const float4*)(W + code * DIMS);
        float s = 0.0f;
#pragma unroll
        for (int j = 0; j < DIMS / 4; ++j) {
            float4 v = wr[j];
            s += v.x * v.x + v.y * v.y + v.z * v.z + v.w * v.w;
        }
        Nl[code] = 0.5f * s;
    }

#if VQ_ASYNC_STAGE
#if __has_builtin(__builtin_amdgcn_s_wait_asynccnt)
    __builtin_amdgcn_s_wait_asynccnt(0);
#else
    asm volatile("s_wait_asynccnt 0x0" ::: "memory");
#endif
#endif
    __syncthreads();

    const int lane = t & 31;
    const int wave = t >> 5;
    const int row_base = blockIdx.x * ROWS_PER_BLOCK + wave * ROWS_PER_WAVE;
    const int n    = lane & 15;   // column (code-within-tile) / row-within-16
    const int hi   = lane >> 4;   // half-wave select
    const int koff = hi * 2;      // K offset inside each 4-wide slice

    // A tile: x[row_base + n][*], 8 slices of K=4, v2f per lane per slice.
    // V_WMMA_F32_16X16X4_F32 A (16x4): lanes 0-15 -> K={0,1}, lanes 16-31 ->
    // K={2,3}; row M = lane%16.
    v2f a[8];
    {
        const float* __restrict__ xrow = x + (size_t)(row_base + n) * DIMS + koff;
#pragma unroll
        for (int s = 0; s < 8; ++s)
            a[s] = *(const v2f*)(xrow + 4 * s);   // 8-byte aligned -> b64 loads
    }

    // Running argmax of score = dot(x_m, w_k) - 0.5*||w_k||^2.
    // C/D layout: VGPR r, lane L -> row M = r + 8*hi, col N = L%16.
    float best[8];
    int   bidx[8];
#pragma unroll
    for (int r = 0; r < 8; ++r) { best[r] = -3.402823466e38f; bidx[r] = 0; }

    for (int k0 = 0; k0 < K_CODES; k0 += 16) {
        // B tile (4x16) mirrors the A layout: lane L supplies column n = L%16,
        // i.e. code k0+n, dims {4s+koff, 4s+koff+1}.
        const float* __restrict__ wrow = Wl + (k0 + n) * DIMS + koff;
        const float wn = Nl[k0 + n];          // one ds_load_b32 per tile
        v8f c;
#pragma unroll
        for (int r = 0; r < 8; ++r) c[r] = -wn;   // fold -0.5||w||^2 into C init
#pragma unroll
        for (int s = 0; s < 8; ++s) {
            v2f b = *(const v2f*)(wrow + 4 * s);  // ds_load_2addr_b64 pairs
            // (neg_a, A, neg_b, B, c_mod, C, reuse_a, reuse_b)
            c = __builtin_amdgcn_wmma_f32_16x16x4_f32(
                    false, a[s], false, b, (short)0, c, false, false);
        }
#pragma unroll
        for (int r = 0; r < 8; ++r) {
            if (c[r] > best[r]) { best[r] = c[r]; bidx[r] = k0 + n; }  // strict > keeps first index
        }
    }

    // Argmax reduction across the 16 columns of each half-wave (first-index tie-break).
#pragma unroll
    for (int off = 1; off < 16; off <<= 1) {
#pragma unroll
        for (int r = 0; r < 8; ++r) {
            float ov = __shfl_xor(best[r], off);
            int   oi = __shfl_xor(bidx[r], off);
            if (ov > best[r] || (ov == best[r] && oi < bidx[r])) {
                best[r] = ov; bidx[r] = oi;
            }
        }
    }
    // Lanes 0-15 hold winners for rows 0..7 (reg r), lanes 16-31 for rows 8..15.

    // Epilogue: gather q = W[idx], fused loss partial, idx output.
    float lsum = 0.0f;
#pragma unroll
    for (int m = 0; m < 16; ++m) {
        const int src  = (m < 8) ? 0 : 16;
        const int idxm = __shfl(bidx[m & 7], src);
        const int row  = row_base + m;
        const float wv = Wl[idxm * DIMS + lane];                 // 32 dims across 32 lanes
        const float xv = x[(size_t)row * DIMS + lane];
        out[(size_t)row * DIMS + lane] = wv;                     // q_ste forward value == q
        const float d = wv - xv;
        lsum += d * d;
        if (lane == 0) out[IDX_OFF + row] = (float)idxm;
    }

    // loss = (BETA + 1) * mean((q - x)^2) = 1.25 * sum / QN
    lsum += __shfl_xor(lsum, 1);
    lsum += __shfl_xor(lsum, 2);
    lsum += __shfl_xor(lsum, 4);
    lsum += __shfl_xor(lsum, 8);
    lsum += __shfl_xor(lsum, 16);
    if (lane == 0)
        atomicAdd(&out[LOSS_OFF], lsum * (1.25f / (float)QN));
}

extern "C" void kernel_launch(void* const* d_in, const int* in_sizes, int n_in,
                              void* d_out, int out_size, void* d_ws, size_t ws_size,
                              hipStream_t stream) {
    const float* x = (const float*)d_in[0];   // [128*32*32, 32] fp32
    const float* W = (const float*)d_in[1];   // [1024, 32] fp32
    float* out = (float*)d_out;               // [q | loss | idx]

    vq_zero_loss<<<1, 1, 0, stream>>>(out);
    vq_argmin_wmma<<<N_ROWS / ROWS_PER_BLOCK, 256, 0, stream>>>(x, W, out);
}